// prms_marrmot_17437567222011
// MI455X (gfx1250) — compile-verified
//
#include <hip/hip_runtime.h>
#include <hip/hip_bf16.h>
#include <stdint.h>

#define NSTEP  1461
#define NGRID  4096
#define NMULV  4
#define UH     15
#define NEARZ  1e-5f

constexpr int GPB = 64;                  // grids per scan block (256 thr / 4 mul)
constexpr int TT  = 16;                  // timesteps per LDS tile
constexpr int NT  = (NSTEP + TT - 1)/TT; // 92 tiles

typedef float v2f __attribute__((ext_vector_type(2)));
typedef float v8f __attribute__((ext_vector_type(8)));

// ---------------------------------------------------------------------------
// Async bulk copy: one 16-step x-tile (16 x 64 grids x 3 comps = 768 b128
// chunks) staged into LDS via the CDNA5 async-to-LDS path (ASYNCcnt).
// ---------------------------------------------------------------------------
__device__ __forceinline__ void issue_tile(const char* xg, int t0, int g0,
                                           uint32_t lds_byte_base, int tid) {
#pragma unroll
  for (int j = 0; j < 3; ++j) {
    int c   = tid + 256 * j;          // chunk id in [0,768)
    int row = c / 48;                 // 48 b128 chunks per timestep row
    int cm  = c % 48;
    int tr  = t0 + row; if (tr > NSTEP - 1) tr = NSTEP - 1;   // clamp tail
    const char* gp = xg + (size_t)tr * (size_t)(NGRID * 12)
                        + (size_t)g0 * 12 + (size_t)cm * 16;
    uint32_t lb = lds_byte_base + (uint32_t)c * 16u;
    asm volatile("global_load_async_to_lds_b128 %0, %1, off"
                 :: "v"(lb), "v"(gp) : "memory");
  }
}

// ---------------------------------------------------------------------------
// Kernel 1: the 1461-step PRMS recurrence. One lane per (grid, mul) chain;
// quad shuffle-reduce produces rf[t][g] = mean_mul(Q).
// ---------------------------------------------------------------------------
__global__ __launch_bounds__(256)
void prms_scan_kernel(const float* __restrict__ x,
                      const float* __restrict__ praw,
                      float* __restrict__ rf) {
  __shared__ __align__(16) float xs[2][TT * GPB * 3];   // 2 x 12 KB

  const int tid = threadIdx.x;
  const int mul = tid & 3;
  const int gl  = tid >> 2;                 // local grid 0..63
  const int g0  = blockIdx.x * GPB;
  const int g   = g0 + gl;

  // ---- params: last timestep slice only, scaled to PARAM_BOUNDS ----
  const size_t pbase = (size_t)(NSTEP - 1) * NGRID * 72 + (size_t)g * 72 + mul;
  float pr[18];
#pragma unroll
  for (int i = 0; i < 18; ++i) pr[i] = praw[pbase + (size_t)i * 4];

  const float tt     = pr[0] * 8.f - 3.f;
  const float ddf    = pr[1] * 20.f;
  const float alpha  = pr[2];
  const float beta   = pr[3];
  const float stor   = pr[4] * 5.f;
  const float retip  = pr[5] * 50.f;
  const float fscn   = pr[6];
  const float scx    = pr[7];
  const float flz    = pr[8] * 0.99f + 0.005f;
  const float stot   = pr[9] * 1999.f + 1.f;
  const float cgw    = pr[10] * 20.f;
  const float resmax = pr[11] * 299.f + 1.f;
  const float k1     = pr[12];
  const float k2     = pr[13] * 4.f + 1.f;
  const float k3     = pr[14];
  const float k4     = pr[15];
  const float k5     = pr[16];
  const float k6     = pr[17];

  const float scn        = fscn * scx;
  const float remx       = (1.f - flz) * stot;
  const float smax       = flz * stot;
  const float inv_remx   = 1.f / remx;
  const float inv_smax   = 1.f / smax;
  const float inv_resmax = 1.f / resmax;
  const float omb        = 1.f - beta;

  float snow = 1e-3f, xin = 1e-3f, rstor = 1e-3f, rechr = 1e-3f;
  float smav = 1e-3f, res = 1e-3f, gw = 1e-3f;

  const char* xg = (const char*)x;
  issue_tile(xg, 0, g0, (uint32_t)(uintptr_t)&xs[0][0], tid);

  for (int tile = 0; tile < NT; ++tile) {
    if (tile + 1 < NT) {
      issue_tile(xg, (tile + 1) * TT, g0,
                 (uint32_t)(uintptr_t)&xs[(tile + 1) & 1][0], tid);
      asm volatile("s_wait_asynccnt 3" ::: "memory");  // current tile landed
    } else {
      asm volatile("s_wait_asynccnt 0" ::: "memory");
    }
    __syncthreads();

    const float* xb = xs[tile & 1];
    const int t0   = tile * TT;
    int tend = NSTEP - t0; if (tend > TT) tend = TT;

    for (int r = 0; r < tend; ++r) {
      const float Pt   = xb[r * (GPB * 3) + gl * 3 + 0];
      const float Tt   = xb[r * (GPB * 3) + gl * 3 + 1];
      const float PETt = xb[r * (GPB * 3) + gl * 3 + 2];

      float flux_ps = (Tt <= tt) ? Pt : 0.f;
      float flux_pr = Pt - flux_ps;
      snow += flux_ps;
      float flux_m = fminf(fmaxf(ddf * (Tt - tt), 0.f), snow);
      snow = fmaxf(snow - flux_m, NEARZ);
      float flux_pim = flux_pr * omb;
      float flux_psm = flux_pr * beta;
      float flux_pby = flux_psm * (1.f - alpha);
      float flux_pin = flux_psm * alpha;
      float flux_mim = flux_m * omb;
      float flux_msm = flux_m * beta;
      xin += flux_pin;
      float flux_ptf = fmaxf(xin - stor, 0.f);
      xin = fmaxf(xin - flux_ptf, NEARZ);
      float flux_ein = fminf(beta * PETt, xin);
      xin = fmaxf(xin - flux_ein, NEARZ);
      rstor += flux_pim + flux_mim;
      float flux_sas = fmaxf(rstor - retip, 0.f);
      rstor = fmaxf(rstor - flux_sas, NEARZ);
      float flux_eim = fminf(omb * PETt, rstor);
      rstor = fmaxf(rstor - flux_eim, NEARZ);
      float sro = scn + (scx - scn) * (rechr * inv_remx);
      sro = fminf(fmaxf(sro, 0.f), 1.f);
      float flux_infil = flux_msm + flux_ptf + flux_pby;
      float flux_sro = sro * flux_infil;
      rechr += flux_infil - flux_sro;
      float flux_pc = fmaxf(rechr - remx, 0.f);
      rechr -= flux_pc;
      float pet_rem = fmaxf(PETt - flux_ein - flux_eim, 0.f);
      float flux_ea = fminf(fmaxf(rechr * inv_remx * pet_rem, 0.f), rechr);
      rechr = fmaxf(rechr - flux_ea, NEARZ);
      smav += flux_pc;
      float flux_excs = fmaxf(smav - smax, 0.f);
      smav -= flux_excs;
      float transp = (rechr < pet_rem)
                       ? fmaxf(smav * inv_smax * (pet_rem - flux_ea), 0.f) : 0.f;
      float flux_et = fminf(transp, smav);
      smav = fmaxf(smav - flux_et, NEARZ);
      float flux_sep  = fminf(cgw, flux_excs);
      float flux_qres = fmaxf(flux_excs - flux_sep, 0.f);
      res += flux_qres;
      float flux_gad = fminf(k1 * __powf(res * inv_resmax, k2), res);
      res = fmaxf(res - flux_gad, NEARZ);
      float flux_ras = fminf(k3 * res + k4 * res * res, res);
      res = fmaxf(res - flux_ras, NEARZ);
      gw += flux_gad + flux_sep;
      float flux_bas = k5 * gw;
      gw = fmaxf(gw - flux_bas, NEARZ);
      float flux_snk = k6 * gw;
      gw = fmaxf(gw - flux_snk, NEARZ);
      float Q = flux_sas + flux_sro + flux_bas + flux_ras;

      // mean over NMUL: butterfly across the 4-lane quad
      float qs = Q + __shfl_xor(Q, 1, 32);
      qs += __shfl_xor(qs, 2, 32);
      if (mul == 0) rf[(size_t)(t0 + r) * NGRID + g] = 0.25f * qs;
    }
    __syncthreads();   // protect buffer being re-filled next iteration
  }
}

// ---------------------------------------------------------------------------
// Kernel 2: normalized gamma unit-hydrograph weights, w[g][0..14], w[g][15]=0.
// ---------------------------------------------------------------------------
__global__ void uh_weights_kernel(const float* __restrict__ conv,
                                  float* __restrict__ w) {
  int g = blockIdx.x * blockDim.x + threadIdx.x;
  if (g >= NGRID) return;
  float a  = conv[g * 2 + 0] * 2.9f;
  float b  = conv[g * 2 + 1] * 6.5f;
  float aa = fmaxf(a, 0.f) + 0.1f;
  float th = fmaxf(b, 0.f) + 0.5f;
  float denom = expf(lgammaf(aa)) * powf(th, aa);
  float wv[UH];
  float s = 0.f;
#pragma unroll
  for (int k = 0; k < UH; ++k) {
    float t = (float)k + 0.5f;
    wv[k] = powf(t, aa - 1.f) * expf(-t / th) / denom;
    s += wv[k];
  }
  float inv = 1.f / s;
#pragma unroll
  for (int k = 0; k < UH; ++k) w[g * 16 + k] = wv[k] * inv;
  w[g * 16 + 15] = 0.f;
}

// ---------------------------------------------------------------------------
// Kernel 3: causal 15-tap routing as fp32 WMMA. One wave per (grid, 16-step
// tile): D = sum_{c=0..3} A_c x B_c with V_WMMA_F32_16X16X4_F32, where
// A[i][k] = rf[g, t0+i-k] (Toeplitz) and B rows are the gamma weights.
// Lane 0 holds out rows M=0..7 (N=0), lane 16 holds M=8..15.
// ---------------------------------------------------------------------------
__global__ __launch_bounds__(256)
void conv_wmma_kernel(const float* __restrict__ rf,
                      const float* __restrict__ w,
                      float* __restrict__ out) {
  const int lane = threadIdx.x & 31;
  const int wid  = blockIdx.x * (blockDim.x >> 5) + (threadIdx.x >> 5);
  if (wid >= NGRID * NT) return;              // wave-uniform exit (EXEC stays full)

  const int g  = wid / NT;
  const int t0 = (wid % NT) * TT;
  const int m  = lane & 15;                   // output row M
  const int h  = (lane < 16) ? 0 : 2;         // K-half select per ISA layout

  v8f acc = {};
#pragma unroll
  for (int c = 0; c < 4; ++c) {
    const int k0 = 4 * c + h;                 // K for VGPR0; VGPR1 is k0+1
    int ta = t0 + m - k0;
    int tb = ta - 1;
    int tac = ta > (NSTEP - 1) ? (NSTEP - 1) : ta;   // clamp (rows discarded)
    int tbc = tb > (NSTEP - 1) ? (NSTEP - 1) : tb;
    v2f a, b;
    a.x = (ta >= 0) ? rf[(size_t)tac * NGRID + g] : 0.f;
    a.y = (tb >= 0) ? rf[(size_t)tbc * NGRID + g] : 0.f;
    b.x = w[g * 16 + k0];
    b.y = w[g * 16 + k0 + 1];
    acc = __builtin_amdgcn_wmma_f32_16x16x4_f32(
        /*neg_a=*/false, a, /*neg_b=*/false, b,
        /*c_mod=*/(short)0, acc, /*reuse_a=*/false, /*reuse_b=*/false);
  }

  if (lane == 0) {
#pragma unroll
    for (int r = 0; r < 8; ++r) {
      int t = t0 + r;
      if (t < NSTEP) out[(size_t)t * NGRID + g] = acc[r];
    }
  } else if (lane == 16) {
#pragma unroll
    for (int r = 0; r < 8; ++r) {
      int t = t0 + 8 + r;
      if (t < NSTEP) out[(size_t)t * NGRID + g] = acc[r];
    }
  }
}

// ---------------------------------------------------------------------------
extern "C" void kernel_launch(void* const* d_in, const int* in_sizes, int n_in,
                              void* d_out, int out_size, void* d_ws, size_t ws_size,
                              hipStream_t stream) {
  const float* x    = (const float*)d_in[0];   // (1461,4096,3)
  // d_in[1] = c_hydro_model: unused by the reference
  const float* praw = (const float*)d_in[2];   // (1461,4096,18,4)
  const float* conv = (const float*)d_in[3];   // (4096,2)
  float* out = (float*)d_out;                  // (1461,4096,1)

  float* rf = (float*)d_ws;                       // NSTEP*NGRID floats (~24 MB)
  float* w  = rf + (size_t)NSTEP * NGRID;         // NGRID*16 floats

  prms_scan_kernel<<<NGRID / GPB, 256, 0, stream>>>(x, praw, rf);
  uh_weights_kernel<<<(NGRID + 255) / 256, 256, 0, stream>>>(conv, w);

  const int nwaves = NGRID * NT;
  const int wpb = 256 / 32;
  conv_wmma_kernel<<<(nwaves + wpb - 1) / wpb, 256, 0, stream>>>(rf, w, out);
}